// GIN_90366111908652
// MI455X (gfx1250) — compile-verified
//
#include <hip/hip_runtime.h>

#define N_NODES 100000
#define N_EDGES 1600000
#define D 128

typedef __attribute__((ext_vector_type(2))) float v2f;
typedef __attribute__((ext_vector_type(8))) float v8f;

// ===================== CSR build (once per launch; edges fixed) ============
__global__ void csr_zero(int* __restrict__ cnt) {
  int i = blockIdx.x * blockDim.x + threadIdx.x;
  if (i < N_NODES) cnt[i] = 0;
}

__global__ void csr_hist(const long long* __restrict__ dst, int* __restrict__ cnt) {
  int e = blockIdx.x * blockDim.x + threadIdx.x;
  if (e < N_EDGES) atomicAdd(&cnt[(int)dst[e]], 1);
}

// Single-block exclusive scan over 100000 counts -> rowptr[N+1]; also
// initializes cursor (in place over cnt) to the row start offsets.
__global__ __launch_bounds__(1024) void csr_scan(int* __restrict__ cnt /*in:count, out:cursor*/,
                                                 int* __restrict__ rowptr) {
  __shared__ int part[1024];
  const int tid = threadIdx.x;
  const int CHUNK = (N_NODES + 1023) / 1024;  // 98
  const int base = tid * CHUNK;
  int s = 0;
  for (int i = 0; i < CHUNK; ++i) {
    int idx = base + i;
    if (idx < N_NODES) s += cnt[idx];
  }
  part[tid] = s;
  __syncthreads();
  // Hillis-Steele inclusive scan of per-thread sums
  for (int off = 1; off < 1024; off <<= 1) {
    int v = (tid >= off) ? part[tid - off] : 0;
    __syncthreads();
    part[tid] += v;
    __syncthreads();
  }
  int run = (tid == 0) ? 0 : part[tid - 1];  // exclusive base for this thread
  for (int i = 0; i < CHUNK; ++i) {
    int idx = base + i;
    if (idx < N_NODES) {
      int c = cnt[idx];       // read count BEFORE overwriting (aliased)
      rowptr[idx] = run;
      cnt[idx] = run;         // becomes the running cursor for csr_fill
      run += c;
    }
  }
  if (tid == 1023) rowptr[N_NODES] = part[1023];  // total = N_EDGES
}

__global__ void csr_fill(const long long* __restrict__ src,
                         const long long* __restrict__ dst,
                         int* __restrict__ cursor, int* __restrict__ col) {
  int e = blockIdx.x * blockDim.x + threadIdx.x;
  if (e < N_EDGES) {
    int d = (int)dst[e];
    int pos = atomicAdd(&cursor[d], 1);
    col[pos] = (int)src[e];
  }
}

// ========= aggregation: pre[i] = (1+eps)*h[i] + sum_{j in N(i)} h[j] =======
// One wave32 per node; each lane owns 4 consecutive features (float4).
// Atomic-free gather-reduce; h (51.2MB) is L2-resident on MI455X (192MB L2).
__global__ void gin_aggregate(const float* __restrict__ h,
                              const float* __restrict__ eps,
                              const int* __restrict__ rowptr,
                              const int* __restrict__ col,
                              float* __restrict__ out) {
  int node = blockIdx.x * (blockDim.x >> 5) + (threadIdx.x >> 5);
  int lane = threadIdx.x & 31;
  if (node >= N_NODES) return;
  const float4* hv = (const float4*)h;
  float s = 1.0f + *eps;
  float4 v = hv[(size_t)node * (D / 4) + lane];
  float4 acc = make_float4(s * v.x, s * v.y, s * v.z, s * v.w);
  int j = rowptr[node];
  const int end = rowptr[node + 1];
  for (; j + 1 < end; j += 2) {  // 2-way unroll for load-latency overlap
    int n0 = col[j], n1 = col[j + 1];
    float4 u0 = hv[(size_t)n0 * (D / 4) + lane];
    float4 u1 = hv[(size_t)n1 * (D / 4) + lane];
    acc.x += u0.x; acc.y += u0.y; acc.z += u0.z; acc.w += u0.w;
    acc.x += u1.x; acc.y += u1.y; acc.z += u1.z; acc.w += u1.w;
  }
  if (j < end) {
    int n0 = col[j];
    float4 u0 = hv[(size_t)n0 * (D / 4) + lane];
    acc.x += u0.x; acc.y += u0.y; acc.z += u0.z; acc.w += u0.w;
  }
  ((float4*)out)[(size_t)node * (D / 4) + lane] = acc;
}

// ========= GEMM: out = relu(pre @ W + b) via V_WMMA_F32_16X16X4_F32 ========
// Block = 128 threads = 4 waves; each wave owns 16 node-rows x all 128 cols
// (8 accumulator tiles, 64 VGPRs of C). W staged in LDS in K-PAIRED layout:
// sW[p*D + n] = {W[2p][n], W[2p+1][n]} so each B fragment is ONE ds_load_b64
// into an even-aligned VGPR pair (no v_mov repacking before v_wmma).
__global__ __launch_bounds__(128) void gin_gemm_relu_wmma(
    const float* __restrict__ H, const float* __restrict__ W,
    const float* __restrict__ bias, float* __restrict__ out) {
  __shared__ float2 sW[(D / 2) * D];  // 64 KB of 320 KB WGP LDS

  for (int i = threadIdx.x; i < (D / 2) * D; i += blockDim.x) {
    int p = i >> 7;       // K-pair index 0..63
    int n = i & (D - 1);  // output column
    float2 pr;
    pr.x = W[(2 * p) * D + n];
    pr.y = W[(2 * p + 1) * D + n];
    sW[i] = pr;
  }
  __syncthreads();

  const int wave = threadIdx.x >> 5;
  const int lane = threadIdx.x & 31;
  const int half = lane >> 4;   // A layout: lanes 0-15 hold K=0..1, 16-31 K=2..3
  const int l15  = lane & 15;
  const int row0 = blockIdx.x * 64 + wave * 16;
  const int myrow = row0 + l15;
  // clamp OOB rows to 0 (valid loads; masked at store; EXEC all-ones for WMMA)
  const float* hrow = H + (size_t)(myrow < N_NODES ? myrow : 0) * D + 2 * half;

  v8f acc[8] = {};

  for (int kk = 0; kk < 32; ++kk) {              // K = 128 = 32 x 4
    v2f a = *(const v2f*)(hrow + 4 * kk);        // A frag (global, L2-resident)
    const v2f* wrow = (const v2f*)(sW + (2 * kk + half) * D + l15);
#pragma unroll
    for (int jt = 0; jt < 8; ++jt) {             // reuse A across 8 col-tiles
      v2f b = wrow[4 * jt];                      // one ds_load_b64: {W[k][n],W[k+1][n]}
      acc[jt] = __builtin_amdgcn_wmma_f32_16x16x4_f32(
          false, a, false, b, (short)0, acc[jt], false, false);
    }
  }

  // epilogue: bias + relu + store. C layout: VGPR r -> M = r + 8*half, N = l15
#pragma unroll
  for (int jt = 0; jt < 8; ++jt) {
    float bb = bias[16 * jt + l15];
#pragma unroll
    for (int r = 0; r < 8; ++r) {
      int row = row0 + r + 8 * half;
      if (row < N_NODES) {
        float v = acc[jt][r] + bb;
        out[(size_t)row * D + 16 * jt + l15] = v > 0.0f ? v : 0.0f;
      }
    }
  }
}

// ============== in-place log_softmax per row, one wave32 per node ==========
__global__ void gin_log_softmax(float* __restrict__ io) {
  int node = blockIdx.x * (blockDim.x >> 5) + (threadIdx.x >> 5);
  int lane = threadIdx.x & 31;
  if (node >= N_NODES) return;
  float4 v = ((const float4*)(io + (size_t)node * D))[lane];
  float m = fmaxf(fmaxf(v.x, v.y), fmaxf(v.z, v.w));
  for (int off = 16; off > 0; off >>= 1) m = fmaxf(m, __shfl_xor(m, off, 32));
  float s = expf(v.x - m) + expf(v.y - m) + expf(v.z - m) + expf(v.w - m);
  for (int off = 16; off > 0; off >>= 1) s += __shfl_xor(s, off, 32);
  float lse = m + logf(s);
  v.x -= lse; v.y -= lse; v.z -= lse; v.w -= lse;
  ((float4*)(io + (size_t)node * D))[lane] = v;
}

extern "C" void kernel_launch(void* const* d_in, const int* in_sizes, int n_in,
                              void* d_out, int out_size, void* d_ws, size_t ws_size,
                              hipStream_t stream) {
  const float* x  = (const float*)d_in[0];
  const long long* edge = (const long long*)d_in[1];  // int64 [2, N_EDGES]
  const float* W1 = (const float*)d_in[2];
  const float* b1 = (const float*)d_in[3];
  const float* e1 = (const float*)d_in[4];
  const float* W2 = (const float*)d_in[5];
  const float* b2 = (const float*)d_in[6];
  const float* e2 = (const float*)d_in[7];
  const float* W3 = (const float*)d_in[8];
  const float* b3 = (const float*)d_in[9];
  const float* e3 = (const float*)d_in[10];

  float* out = (float*)d_out;  // doubles as h buffer between layers

  // workspace layout
  float* pre    = (float*)d_ws;                       // N*D f32 = 51.2 MB
  int*   rowptr = (int*)(pre + (size_t)N_NODES * D);  // N+1 ints
  int*   cursor = rowptr + (N_NODES + 1);             // N ints (count->cursor)
  int*   col    = cursor + N_NODES;                   // N_EDGES ints

  const long long* src = edge;
  const long long* dst = edge + N_EDGES;

  const int node_blocks    = (N_NODES + 255) / 256;
  const int edge_blocks    = (N_EDGES + 255) / 256;
  const int wave_blocks    = (N_NODES + 7) / 8;   // 8 waves of 32 per block
  const int gemm_blocks    = (N_NODES + 63) / 64;

  // ---- build CSR by destination (deterministic work; edges are constant) ----
  csr_zero <<<node_blocks, 256, 0, stream>>>(cursor);
  csr_hist <<<edge_blocks, 256, 0, stream>>>(dst, cursor);
  csr_scan <<<1, 1024, 0, stream>>>(cursor, rowptr);
  csr_fill <<<edge_blocks, 256, 0, stream>>>(src, dst, cursor, col);

  // ---- layer 1 ----
  gin_aggregate     <<<wave_blocks, 256, 0, stream>>>(x,   e1, rowptr, col, pre);
  gin_gemm_relu_wmma<<<gemm_blocks, 128, 0, stream>>>(pre, W1, b1, out);
  // ---- layer 2 ----
  gin_aggregate     <<<wave_blocks, 256, 0, stream>>>(out, e2, rowptr, col, pre);
  gin_gemm_relu_wmma<<<gemm_blocks, 128, 0, stream>>>(pre, W2, b2, out);
  // ---- layer 3 ----
  gin_aggregate     <<<wave_blocks, 256, 0, stream>>>(out, e3, rowptr, col, pre);
  gin_gemm_relu_wmma<<<gemm_blocks, 128, 0, stream>>>(pre, W3, b3, out);
  // ---- log_softmax (in place, wave-private rows) ----
  gin_log_softmax   <<<wave_blocks, 256, 0, stream>>>(out);
}